// MorphologicalOpeningTransform_2319282340048
// MI455X (gfx1250) — compile-verified
//
#include <hip/hip_runtime.h>
#include <stdint.h>

// Morphological opening (erode-then-dilate, 15x15 ellipse) for 3x2048x2048 fp32.
//
// MI455X reasoning: memory floor = 2 passes * (48MB read + 48MB write) = 192MB
// -> ~8.2us @ 23.3 TB/s. Naive footprint reduction is 169 VALU min/max per
// pixel per pass (VALU-bound, ~5x off roofline). We decompose the ellipse into
// row-runs: result = vertical min over horizontal mins of widths {1,9,11,13,15}
// (widths share prefixes: 8+2+2+2 ops), streamed through register ring
// accumulators -> ~45 VALU ops + ~29 LDS dwords per pixel per pass, which puts
// the kernel back at the HBM roofline. WMMA does not apply (no matmul
// structure); the CDNA5 feature that pays here is the async global->LDS copy
// path (global_load_async_to_lds_b128 + s_wait_asynccnt) for halo-tile staging.

#define IMG_W 2048
#define IMG_H 2048
#define IMG_C 3
#define TW 64            // tile width  (output cols per workgroup)
#define TH 64            // tile height (output rows per workgroup)
#define SROWS 16         // output rows per thread strip (TH / blockDim.y)
#define RW 80            // LDS raw row stride in floats (16B-aligned rows)
#define RH (TH + 14)     // 78 raw rows (7-halo each side)

template <bool ERODE>
__device__ __forceinline__ float redop(float a, float b) {
  return ERODE ? fminf(a, b) : fmaxf(a, b);
}

template <bool ERODE>
__global__ __launch_bounds__(256)
void morph_pass(const float* __restrict__ gin, float* __restrict__ gout) {
  __shared__ float raw[RH * RW];           // 24960 B

  const int tx  = threadIdx.x;             // 0..63  (one output column each)
  const int ty  = threadIdx.y;             // 0..3   (row-strip id)
  const int tid = ty * TW + tx;
  const int gx0 = blockIdx.x * TW;
  const int gy0 = blockIdx.y * TH;
  const int ch  = blockIdx.z;
  const float PAD = ERODE ? __builtin_inff() : -__builtin_inff();
  const float* gch = gin + (size_t)ch * IMG_W * IMG_H;

  // ---------------- Phase 1: stage 80x78 halo tile into LDS ----------------
  // Interior tiles: async DMA-style copy, rows start at gx0-8 (16B aligned).
  const bool interior = (gx0 >= 8) && (gx0 + 72 <= IMG_W) &&
                        (gy0 >= 7) && (gy0 + TH + 7 <= IMG_H);
  if (interior) {
    const float* src = gch + (size_t)(gy0 - 7) * IMG_W + (gx0 - 8);
    for (int u = tid; u < RH * 20; u += 256) {         // 20 x b128 per row
      const int r  = u / 20;
      const int cc = u % 20;
      const float* ga = src + (size_t)r * IMG_W + cc * 4;
      const uint32_t lo = (uint32_t)(uintptr_t)&raw[r * RW + cc * 4];
      asm volatile("global_load_async_to_lds_b128 %0, %1, off"
                   :: "v"(lo), "v"(ga) : "memory");
    }
    asm volatile("s_wait_asynccnt 0" ::: "memory");
  } else {
    // Border tiles: per-element load with +/-inf pad (matches reference pad).
    for (int u = tid; u < RH * RW; u += 256) {
      const int r  = u / RW;
      const int xi = u % RW;
      const int gy = gy0 - 7 + r;
      const int gx = gx0 - 8 + xi;
      float v = PAD;
      if (gy >= 0 && gy < IMG_H && gx >= 0 && gx < IMG_W)
        v = gch[(size_t)gy * IMG_W + gx];
      raw[u] = v;
    }
  }
  __syncthreads();

  // ---------------- Phase 2: row-run decomposition + vertical ring ---------
  // Ellipse row half-widths by |dy|: {0,1,2}->7, {3,4}->6, {5}->5, {6}->4,
  // {7}->0  => widths 15,15,15,13,13,11,9,1.
  float acc[SROWS];
#pragma unroll
  for (int k = 0; k < SROWS; ++k) acc[k] = PAD;

  const int rbase = ty * SROWS;
#pragma unroll
  for (int p = 0; p < SROWS + 14; ++p) {
    const float* row = &raw[(rbase + p) * RW + tx];
    float v[15];                       // raw cols tx+1 .. tx+15 (center tx+8)
#pragma unroll
    for (int j = 0; j < 15; ++j) v[j] = row[j + 1];

    float h9 = v[3];                   // offsets -4..+4
#pragma unroll
    for (int j = 4; j <= 11; ++j) h9 = redop<ERODE>(h9, v[j]);
    const float h11 = redop<ERODE>(h9,  redop<ERODE>(v[2], v[12]));
    const float h13 = redop<ERODE>(h11, redop<ERODE>(v[1], v[13]));
    const float h15 = redop<ERODE>(h13, redop<ERODE>(v[0], v[14]));
    const float c7  = v[7];            // width-1 row (|dy| == 7)

#pragma unroll
    for (int k = 0; k < SROWS; ++k) {  // raw row p feeds outputs k = p-14..p
      const int d = p - k;
      if (d < 0 || d > 14) continue;
      const int ady = d < 7 ? 7 - d : d - 7;     // |dy| (compile-time)
      const float contrib = (ady == 7) ? c7
                          : (ady == 6) ? h9
                          : (ady == 5) ? h11
                          : (ady >= 3) ? h13
                          : h15;
      acc[k] = redop<ERODE>(acc[k], contrib);
    }
  }

  float* och = gout + (size_t)ch * IMG_W * IMG_H;
#pragma unroll
  for (int k = 0; k < SROWS; ++k) {
    och[(size_t)(gy0 + rbase + k) * IMG_W + (gx0 + tx)] = acc[k];
  }
}

extern "C" void kernel_launch(void* const* d_in, const int* in_sizes, int n_in,
                              void* d_out, int out_size, void* d_ws, size_t ws_size,
                              hipStream_t stream) {
  (void)in_sizes; (void)n_in; (void)out_size; (void)ws_size;
  const float* img = (const float*)d_in[0];
  float* out = (float*)d_out;
  float* tmp = (float*)d_ws;                 // needs 3*2048*2048*4 = 48 MiB

  dim3 grid(IMG_W / TW, IMG_H / TH, IMG_C);  // 32 x 32 x 3 workgroups
  dim3 block(TW, 4, 1);                      // 256 threads = 8 wave32

  morph_pass<true ><<<grid, block, 0, stream>>>(img, tmp);  // erosion (+inf pad)
  morph_pass<false><<<grid, block, 0, stream>>>(tmp, out);  // dilation (-inf pad)
}